// TimePredictor_16466904613093
// MI455X (gfx1250) — compile-verified
//
#include <hip/hip_runtime.h>

// TimePredictor fused GRU — MI455X (gfx1250, wave32)
//
// Kernel 1: gx_all[B*T, 96] = concat(emb,ctx) @ W_ih^T + b_ih  (f32 WMMA 16x16x4)
// Kernel 2: per-16-batch-row wave: 512-step GRU scan (f32 WMMA for h@W_hh^T) + head
//
// Workspace: gx_all = B*T*96 floats = 384 MB in d_ws.

typedef __attribute__((ext_vector_type(2))) float v2f;
typedef __attribute__((ext_vector_type(8))) float v8f;

#define B_  2048
#define T_  512
#define E_  128
#define C_  64
#define D_  192   // E + C
#define H_  32
#define G_  96    // 3H

__device__ __forceinline__ float sigmoid_f(float x) {
    return 1.0f / (1.0f + __expf(-x));
}
__device__ __forceinline__ float tanh_f(float x) {
    // tanh(x) = 1 - 2/(exp(2x)+1)
    return 1.0f - 2.0f / (__expf(2.0f * x) + 1.0f);
}

// -------------------------------------------------------------------------
// Kernel 1: gate projection GEMM. One wave -> 16 rows x 96 cols output tile.
// 8 waves per block; W_ih^T staged in dynamic LDS as K-pair-major float2s:
//   sW[(p*G + n)*2 + i] = W_ih[n*D + 2p + i]   (p = k/2)
// so a B-fragment (K rows {k0,k0+1 | k0+2,k0+3} x 16 cols) is one b64 load.
// -------------------------------------------------------------------------
__global__ __launch_bounds__(256) void gates_gemm_kernel(
    const float* __restrict__ emb, const float* __restrict__ ctx,
    const float* __restrict__ W_ih, const float* __restrict__ b_ih,
    float* __restrict__ gx)
{
    extern __shared__ float sW[];  // (D/2)*G*2 floats = 73728 B

    for (int idx = threadIdx.x; idx < (D_ / 2) * G_; idx += 256) {
        int p = idx / G_, n = idx % G_;
        sW[idx * 2 + 0] = W_ih[n * D_ + 2 * p + 0];
        sW[idx * 2 + 1] = W_ih[n * D_ + 2 * p + 1];
    }
    __syncthreads();

    const int lane  = threadIdx.x & 31;
    const int l15   = lane & 15;
    const int khalf = lane >> 4;       // 0: K pair {0,1}, 1: K pair {2,3}
    const int koff  = khalf * 2;
    const int mbase = khalf * 8;       // C-layout row base for this half-wave

    const long gwave   = (long)blockIdx.x * 8 + (threadIdx.x >> 5);
    const long rowbase = gwave * 16;
    const long row     = rowbase + l15;   // A-layout: lanes 0-15 & 16-31 both rows 0..15

    const float* erow = emb + row * E_;
    const float* crow = ctx + row * C_;

    // C tile j covers gate cols [16j, 16j+16); bias depends only on column.
    v8f acc[6];
#pragma unroll
    for (int j = 0; j < 6; ++j) {
        float bj = b_ih[j * 16 + l15];
#pragma unroll
        for (int v = 0; v < 8; ++v) acc[j][v] = bj;
    }

    for (int c = 0; c < 48; ++c) {           // K chunks of 4 (48*4 = 192)
        int k = 4 * c + koff;                // chunk never straddles emb/ctx (128%4==0)
        v2f a = (k < E_) ? *(const v2f*)(erow + k)
                         : *(const v2f*)(crow + (k - E_));
        const float* bp = &sW[((k >> 1) * G_) * 2];
#pragma unroll
        for (int j = 0; j < 6; ++j) {
            v2f b = *(const v2f*)(bp + (j * 16 + l15) * 2);
            acc[j] = __builtin_amdgcn_wmma_f32_16x16x4_f32(
                false, a, false, b, (short)0, acc[j], false, false);
        }
    }

    // Store: C layout — lane holds col l15, rows mbase..mbase+7 in VGPRs 0..7.
#pragma unroll
    for (int v = 0; v < 8; ++v) {
        float* orow = gx + (rowbase + mbase + v) * G_;
#pragma unroll
        for (int j = 0; j < 6; ++j)
            orow[j * 16 + l15] = acc[j][v];
    }
}

// -------------------------------------------------------------------------
// Kernel 2: GRU scan. One wave per 16 batch rows. h kept in WMMA C layout
// (2 tiles: hidden cols 0-15, 16-31). Per step: h -> LDS -> A-frags,
// gh = h @ W_hh^T + b_hh via 8x6 WMMA (B-frags register-resident),
// lane-local nonlinear update (gate col & hidden col share lane/VGPR slot).
// -------------------------------------------------------------------------
__global__ __launch_bounds__(32) void gru_scan_kernel(
    const float* __restrict__ gx,
    const float* __restrict__ W_hh, const float* __restrict__ b_hh,
    const float* __restrict__ W1, const float* __restrict__ b1,
    const float* __restrict__ W2, const float* __restrict__ b2,
    float* __restrict__ out)
{
    __shared__ __attribute__((aligned(16))) float sWh[(H_ / 2) * G_ * 2]; // 12 KB
    __shared__ __attribute__((aligned(16))) float sh[16 * 34];            // h staging (pad 34)

    const int lane  = threadIdx.x;
    const int l15   = lane & 15;
    const int khalf = lane >> 4;
    const int koff  = khalf * 2;
    const int mbase = khalf * 8;
    const long rowbase = (long)blockIdx.x * 16;

    // Pack W_hh^T (K-pair-major) into LDS.
    for (int idx = lane; idx < (H_ / 2) * G_; idx += 32) {
        int p = idx / G_, n = idx % G_;
        sWh[idx * 2 + 0] = W_hh[n * H_ + 2 * p + 0];
        sWh[idx * 2 + 1] = W_hh[n * H_ + 2 * p + 1];
    }
    __syncthreads();

    // Hoist all B fragments (invariant across the 512 steps): 8 K-chunks x 6 N-tiles.
    v2f Bf[48];
#pragma unroll
    for (int c = 0; c < 8; ++c) {
        int p = 2 * c + khalf;
#pragma unroll
        for (int j = 0; j < 6; ++j)
            Bf[c * 6 + j] = *(const v2f*)(&sWh[(p * G_ + j * 16 + l15) * 2]);
    }

    float bias[6];
#pragma unroll
    for (int j = 0; j < 6; ++j) bias[j] = b_hh[j * 16 + l15];

    v8f h0, h1;   // hidden cols [0,16) and [16,32) in C layout
#pragma unroll
    for (int v = 0; v < 8; ++v) { h0[v] = 0.0f; h1[v] = 0.0f; }

    for (int t = 0; t < T_; ++t) {
        // Stage h to LDS in plain [batch][hidden] (row stride 34, so k-even
        // addresses stay 8B aligned for the v2f A-frag reads).
#pragma unroll
        for (int v = 0; v < 8; ++v) {
            sh[(mbase + v) * 34 + l15]      = h0[v];
            sh[(mbase + v) * 34 + 16 + l15] = h1[v];
        }
        __syncthreads();

        v8f acc[6];
#pragma unroll
        for (int j = 0; j < 6; ++j) {
#pragma unroll
            for (int v = 0; v < 8; ++v) acc[j][v] = bias[j];
        }

#pragma unroll
        for (int c = 0; c < 8; ++c) {
            int k = 4 * c + koff;
            v2f a = *(const v2f*)(&sh[l15 * 34 + k]);   // row l15, K {k,k+1}
#pragma unroll
            for (int j = 0; j < 6; ++j)
                acc[j] = __builtin_amdgcn_wmma_f32_16x16x4_f32(
                    false, a, false, Bf[c * 6 + j], (short)0, acc[j], false, false);
        }

        // Lane-local GRU update. Tiles: r -> {0,1}, z -> {2,3}, n -> {4,5};
        // hidden unit l15 pairs with tile{0,2,4}, unit 16+l15 with tile{1,3,5}.
#pragma unroll
        for (int v = 0; v < 8; ++v) {
            const float* g = gx + ((rowbase + mbase + v) * T_ + t) * (long)G_;
            float r0 = sigmoid_f(g[ 0 + l15] + acc[0][v]);
            float r1 = sigmoid_f(g[16 + l15] + acc[1][v]);
            float z0 = sigmoid_f(g[32 + l15] + acc[2][v]);
            float z1 = sigmoid_f(g[48 + l15] + acc[3][v]);
            float n0 = tanh_f(g[64 + l15] + r0 * acc[4][v]);
            float n1 = tanh_f(g[80 + l15] + r1 * acc[5][v]);
            h0[v] = (1.0f - z0) * n0 + z0 * h0[v];
            h1[v] = (1.0f - z1) * n1 + z1 * h1[v];
        }
        __syncthreads();   // WAR on sh before next step's stores
    }

    // Head: Linear(32,16) -> ReLU -> Linear(16,1) -> Sigmoid (tiny; lanes 0-15).
#pragma unroll
    for (int v = 0; v < 8; ++v) {
        sh[(mbase + v) * 34 + l15]      = h0[v];
        sh[(mbase + v) * 34 + 16 + l15] = h1[v];
    }
    __syncthreads();
    if (lane < 16) {
        float o = b2[0];
        for (int j = 0; j < 16; ++j) {
            float s = b1[j];
            for (int k = 0; k < H_; ++k) s += sh[lane * 34 + k] * W1[j * H_ + k];
            o += fmaxf(s, 0.0f) * W2[j];
        }
        out[rowbase + lane] = sigmoid_f(o);
    }
}

// -------------------------------------------------------------------------
extern "C" void kernel_launch(void* const* d_in, const int* in_sizes, int n_in,
                              void* d_out, int out_size, void* d_ws, size_t ws_size,
                              hipStream_t stream) {
    const float* emb  = (const float*)d_in[0];
    const float* ctx  = (const float*)d_in[1];
    const float* W_ih = (const float*)d_in[2];
    const float* W_hh = (const float*)d_in[3];
    const float* b_ih = (const float*)d_in[4];
    const float* b_hh = (const float*)d_in[5];
    const float* W1   = (const float*)d_in[6];
    const float* b1   = (const float*)d_in[7];
    const float* W2   = (const float*)d_in[8];
    const float* b2   = (const float*)d_in[9];
    float* out = (float*)d_out;
    float* gx  = (float*)d_ws;   // B*T*G floats = 384 MB

    // 1M rows / 16 rows-per-wave / 8 waves-per-block = 8192 blocks.
    const int blocks1 = (B_ * T_) / 16 / 8;
    const size_t lds1 = (size_t)(D_ / 2) * G_ * 2 * sizeof(float);  // 73728 B
    gates_gemm_kernel<<<blocks1, 256, lds1, stream>>>(emb, ctx, W_ih, b_ih, gx);

    // 2048 batch rows / 16 per wave = 128 single-wave blocks.
    gru_scan_kernel<<<B_ / 16, 32, 0, stream>>>(gx, W_hh, b_hh, W1, b1, W2, b2, out);
}